// MambaSSM_12386685682108
// MI455X (gfx1250) — compile-verified
//
#include <hip/hip_runtime.h>
#include <hip/hip_bf16.h>
#include <math.h>

// ---------------------------------------------------------------------------
// CDNA5 (gfx1250) implementation of the CNN + Mamba-SSM + decoder pipeline.
//  * GEMMs use V_WMMA_F32_16X16X4_F32 (fp32 matrix path, wave32).
//    - strip kernel: 16(M)x64(N) per wave, 4 accumulators, A-fragment reuse.
//  * conv1d weights staged into LDS via the Tensor Data Mover
//    (tensor_load_to_lds + s_wait_tensorcnt), with a DS fallback.
// ---------------------------------------------------------------------------

typedef float v2f __attribute__((ext_vector_type(2)));
typedef float v8f __attribute__((ext_vector_type(8)));

__device__ __forceinline__ float sigmoidf_(float x) { return 1.0f / (1.0f + __expf(-x)); }
__device__ __forceinline__ float siluf_(float x)    { return x * sigmoidf_(x); }

// ===========================================================================
// Tensor Data Mover: 1-D fp32 tile (global -> LDS).  D# packed per ISA
// CDNA5 ISA: group0 {count=1, lds_addr,
// global_addr[56:0], type=2}; group1 {data_size=4B, tensor_dim0=nelem,
// tensor_dim1=1, tile_dim0=nelem, tile_dim1=1, stride0=nelem}.
// This toolchain exposes the 6-arg builtin form:
//   (u32x4 g0, i32x8 g1, i32x4 g2, i32x4 g3, i32x8, i32 cpol)
// ===========================================================================
#if defined(__gfx1250__) && __has_builtin(__builtin_amdgcn_tensor_load_to_lds)
#define USE_TDM 1
typedef unsigned int u32x4 __attribute__((ext_vector_type(4)));
typedef int          i32x4 __attribute__((ext_vector_type(4)));
typedef int          i32x8 __attribute__((ext_vector_type(8)));

__device__ __forceinline__ unsigned lds_offset_of_(const void* p)
{
    // generic -> LDS addrspace cast yields the 32-bit LDS byte offset
    return (unsigned)(unsigned long long)(__attribute__((address_space(3))) char*)(char*)p;
}

__device__ __forceinline__ void tdm_load_f32_1d_(void* lds_dst, const float* gsrc, unsigned nelem)
{
    const unsigned long long ga = (unsigned long long)(const void*)gsrc;
    u32x4 g0;
    g0[0] = 1u;                                               // count=1 (valid user D#)
    g0[1] = lds_offset_of_(lds_dst);                          // lds_addr (bytes)
    g0[2] = (unsigned)ga;                                     // global_addr[31:0]
    g0[3] = (unsigned)((ga >> 32) & 0x1FFFFFFu) | 0x80000000u;// addr[56:32] | type=2

    i32x8 g1;
    g1[0] = (int)(2u << 16);                                  // data_size=2 (4B), wg_mask=0
    g1[1] = (int)((nelem & 0xFFFFu) << 16);                   // tensor_dim0[15:0]
    g1[2] = (int)(((nelem >> 16) & 0xFFFFu) | (1u << 16));    // tensor_dim0[31:16] | tensor_dim1=1
    g1[3] = (int)((nelem & 0xFFFFu) << 16);                   // tensor_dim1 hi=0 | tile_dim0=nelem
    g1[4] = 1;                                                // tile_dim1=1, tile_dim2=0
    g1[5] = (int)nelem;                                       // tensor_dim0_stride[31:0]
    g1[6] = 0;
    g1[7] = 0;

    i32x4 z4 = {};
    i32x8 z8 = {};
    __builtin_amdgcn_tensor_load_to_lds(g0, g1, z4, z4, z8, 0);
}
#else
#define USE_TDM 0
#endif

// ===========================================================================
// GEMM (strip): C[M,N] = A[M,K] @ W[N,K]^T (+bias), N % 64 == 0, K % 4 == 0.
// One wave32 computes a 16x64 strip: one A fragment drives 4 WMMAs per K-step
// (4x A-load reuse, 4 independent matrix ops in flight per loadcnt wait).
// fp32 WMMA layout (ISA 7.12.2):
//   A (16x4):  lanes 0-15 -> M=lane,{K=0,1}; lanes 16-31 -> M=lane-16,{K=2,3}
//   B (4x16):  same pattern with N striped across lanes
//   C (16x16): VGPR i -> M = i + 8*(lane>=16), N = lane&15
// ===========================================================================
__global__ __launch_bounds__(128)
void gemm_wmma_f32_strip_kernel(const float* __restrict__ A, int lda,
                                const float* __restrict__ W,
                                const float* __restrict__ bias,
                                float* __restrict__ C, int ldc,
                                int M, int N, int K)
{
    const int wave    = (blockIdx.x * blockDim.x + threadIdx.x) >> 5;
    const int strips  = N >> 6;
    const int tm      = wave / strips;
    const int tn      = wave - tm * strips;
    if (tm >= (M >> 4)) return;                  // wave-uniform: EXEC stays full

    const int lane = threadIdx.x & 31;
    const int half = lane >> 4;
    const int r    = lane & 15;

    const float* a_ptr = A + (size_t)(tm * 16 + r) * lda + 2 * half;
    const float* w0 = W + (size_t)(tn * 64 +  0 + r) * K + 2 * half;
    const float* w1 = W + (size_t)(tn * 64 + 16 + r) * K + 2 * half;
    const float* w2 = W + (size_t)(tn * 64 + 32 + r) * K + 2 * half;
    const float* w3 = W + (size_t)(tn * 64 + 48 + r) * K + 2 * half;

    v8f acc0 = {}, acc1 = {}, acc2 = {}, acc3 = {};
    #pragma unroll 2
    for (int k = 0; k < K; k += 4) {
        if (k + 32 < K) {
            __builtin_prefetch(a_ptr + k + 32, 0, 3);
            __builtin_prefetch(w0 + k + 32, 0, 3);
            __builtin_prefetch(w2 + k + 32, 0, 3);
        }
        v2f av, b0, b1, b2, b3;
        av.x = a_ptr[k]; av.y = a_ptr[k + 1];
        b0.x = w0[k];    b0.y = w0[k + 1];
        b1.x = w1[k];    b1.y = w1[k + 1];
        b2.x = w2[k];    b2.y = w2[k + 1];
        b3.x = w3[k];    b3.y = w3[k + 1];
        acc0 = __builtin_amdgcn_wmma_f32_16x16x4_f32(false, av, false, b0, (short)0, acc0, false, false);
        acc1 = __builtin_amdgcn_wmma_f32_16x16x4_f32(false, av, false, b1, (short)0, acc1, false, false);
        acc2 = __builtin_amdgcn_wmma_f32_16x16x4_f32(false, av, false, b2, (short)0, acc2, false, false);
        acc3 = __builtin_amdgcn_wmma_f32_16x16x4_f32(false, av, false, b3, (short)0, acc3, false, false);
    }

    #pragma unroll
    for (int j = 0; j < 4; ++j) {
        const v8f acc = (j == 0) ? acc0 : (j == 1) ? acc1 : (j == 2) ? acc2 : acc3;
        const int col = tn * 64 + j * 16 + r;
        const float bv = bias ? bias[col] : 0.0f;
        #pragma unroll
        for (int i = 0; i < 8; ++i) {
            const int row = tm * 16 + half * 8 + i;
            C[(size_t)row * ldc + col] = acc[i] + bv;
        }
    }
}

// Single-tile variant for N % 16 == 0 (used for x_proj, N=144).
__global__ __launch_bounds__(128)
void gemm_wmma_f32_kernel(const float* __restrict__ A, int lda,
                          const float* __restrict__ W,
                          const float* __restrict__ bias,
                          float* __restrict__ C, int ldc,
                          int M, int N, int K)
{
    const int wave     = (blockIdx.x * blockDim.x + threadIdx.x) >> 5;
    const int tn_count = N >> 4;
    const int tm       = wave / tn_count;
    const int tn       = wave - tm * tn_count;
    if (tm >= (M >> 4)) return;

    const int lane = threadIdx.x & 31;
    const int half = lane >> 4;
    const int r    = lane & 15;

    const float* a_ptr = A + (size_t)(tm * 16 + r) * lda + 2 * half;
    const float* w_ptr = W + (size_t)(tn * 16 + r) * K   + 2 * half;

    v8f acc = {};
    #pragma unroll 4
    for (int k = 0; k < K; k += 4) {
        if (k + 32 < K) {
            __builtin_prefetch(a_ptr + k + 32, 0, 3);
            __builtin_prefetch(w_ptr + k + 32, 0, 3);
        }
        v2f av, bv;
        av.x = a_ptr[k]; av.y = a_ptr[k + 1];
        bv.x = w_ptr[k]; bv.y = w_ptr[k + 1];
        acc = __builtin_amdgcn_wmma_f32_16x16x4_f32(false, av, false, bv, (short)0, acc, false, false);
    }

    const int   col = tn * 16 + r;
    const float bv  = bias ? bias[col] : 0.0f;
    #pragma unroll
    for (int i = 0; i < 8; ++i) {
        const int row = tm * 16 + half * 8 + i;
        C[(size_t)row * ldc + col] = acc[i] + bv;
    }
}

// ===========================================================================
// Direct conv2d (also ConvTranspose2d(k3,s1,p1) via flipped/transposed weight
// indexing).  act: 0=none 1=relu 2=sigmoid.
// ===========================================================================
__global__ void conv2d_kernel(const float* __restrict__ in, const float* __restrict__ w,
                              const float* __restrict__ bias, float* __restrict__ out,
                              int B, int Cin, int Hin, int Win,
                              int Cout, int Hout, int Wout,
                              int KH, int KW, int stride, int pad,
                              int act, int deconv)
{
    const int idx   = blockIdx.x * blockDim.x + threadIdx.x;
    const int total = B * Cout * Hout * Wout;
    if (idx >= total) return;

    int t  = idx;
    const int x  = t % Wout; t /= Wout;
    const int y  = t % Hout; t /= Hout;
    const int co = t % Cout;
    const int b  = t / Cout;

    float acc = bias ? bias[co] : 0.0f;
    for (int ci = 0; ci < Cin; ++ci) {
        const float* inp = in + (size_t)(b * Cin + ci) * Hin * Win;
        for (int kh = 0; kh < KH; ++kh) {
            const int iy = y * stride - pad + kh;
            if (iy < 0 || iy >= Hin) continue;
            for (int kw = 0; kw < KW; ++kw) {
                const int ix = x * stride - pad + kw;
                if (ix < 0 || ix >= Win) continue;
                float wv;
                if (deconv)  // w stored (Cin, Cout, KH, KW); flip spatial
                    wv = w[(((size_t)ci * Cout + co) * KH + (KH - 1 - kh)) * KW + (KW - 1 - kw)];
                else         // w stored (Cout, Cin, KH, KW)
                    wv = w[(((size_t)co * Cin + ci) * KH + kh) * KW + kw];
                acc = fmaf(inp[iy * Win + ix], wv, acc);
            }
        }
    }
    if (act == 1)      acc = fmaxf(acc, 0.0f);
    else if (act == 2) acc = sigmoidf_(acc);
    out[idx] = acc;
}

__global__ void add_inplace_kernel(float* __restrict__ out, const float* __restrict__ other, int n)
{
    int i = blockIdx.x * blockDim.x + threadIdx.x;
    if (i < n) out[i] += other[i];
}

// (b,C,HW) -> (b,HW,C)
__global__ void chw_to_lc_kernel(const float* __restrict__ in, float* __restrict__ out,
                                 int B, int C, int HW)
{
    int idx = blockIdx.x * blockDim.x + threadIdx.x;
    int total = B * C * HW;
    if (idx >= total) return;
    int c = idx % C; int t = idx / C;
    int l = t % HW;  int b = t / HW;
    out[idx] = in[((size_t)(b * C + c)) * HW + l];
}

// (b,HW,C) -> (b,C,HW)
__global__ void lc_to_chw_kernel(const float* __restrict__ in, float* __restrict__ out,
                                 int B, int C, int HW)
{
    int idx = blockIdx.x * blockDim.x + threadIdx.x;
    int total = B * C * HW;
    if (idx >= total) return;
    int l = idx % HW; int t = idx / HW;
    int c = t % C;    int b = t / C;
    out[idx] = in[((size_t)(b * HW + l)) * C + c];
}

// ===========================================================================
// Causal depthwise conv1d over sequence + SiLU.  xz is (B*L,1024); xs = cols
// [0,512).  Weight table (512x4 = 8KB) staged into LDS by the Tensor Data
// Mover (wave 0 issues tensor_load_to_lds, waits TENSORcnt, then barrier).
// ===========================================================================
__global__ __launch_bounds__(256)
void conv1d_silu_kernel(const float* __restrict__ xz,
                        const float* __restrict__ w,    // (512,4)
                        const float* __restrict__ bias, // (512)
                        float* __restrict__ u,          // (B*L, 512)
                        int B, int L)
{
    __shared__ float wsh[512 * 4];
    __shared__ float bsh[512];

#if USE_TDM
    if (threadIdx.x < 32) {                      // wave 0: TDM DMA of weight table
        tdm_load_f32_1d_(wsh, w, 512 * 4);
        __builtin_amdgcn_s_wait_tensorcnt(0);
    }
    for (int i = threadIdx.x; i < 512; i += 256) bsh[i] = bias[i];
#else
    for (int i = threadIdx.x; i < 2048; i += 256) wsh[i] = w[i];
    for (int i = threadIdx.x; i < 512;  i += 256) bsh[i] = bias[i];
#endif
    __syncthreads();

    int idx = blockIdx.x * blockDim.x + threadIdx.x;
    int total = B * L * 512;
    if (idx >= total) return;
    int d = idx % 512; int t = idx / 512;
    int l = t % L;     int b = t / L;

    float acc = bsh[d];
    #pragma unroll
    for (int j = 0; j < 4; ++j) {
        int ls = l - 3 + j;
        if (ls < 0) continue;
        acc = fmaf(xz[((size_t)(b * L + ls)) * 1024 + d], wsh[d * 4 + j], acc);
    }
    u[idx] = siluf_(acc);
}

// delta[i] = softplus(delta[i] + bias[d])
__global__ void bias_softplus_kernel(float* __restrict__ delta, const float* __restrict__ bias,
                                     int M, int N)
{
    int idx = blockIdx.x * blockDim.x + threadIdx.x;
    if (idx >= M * N) return;
    float v = delta[idx] + bias[idx % N];
    delta[idx] = (v > 20.0f) ? v : log1pf(__expf(v));
}

__global__ void neg_exp_kernel(const float* __restrict__ in, float* __restrict__ out, int n)
{
    int i = blockIdx.x * blockDim.x + threadIdx.x;
    if (i < n) out[i] = -__expf(in[i]);
}

// ===========================================================================
// Selective-scan: one wave32 per (b,d); each lane owns states n=lane, lane+32.
// h_n <- h_n * exp(dt*A[d,n]) + dt*u*B_t[n];  y = sum_n h_n * C_t[n]
// B_t/C_t are slices of x_dbl (row stride 144, offsets 16 / 80).
// ===========================================================================
__global__ __launch_bounds__(32)
void ssm_scan_kernel(const float* __restrict__ u,     // (B*L,512)
                     const float* __restrict__ delta, // (B*L,512)
                     const float* __restrict__ xdbl,  // (B*L,144)
                     const float* __restrict__ Aneg,  // (512,64)
                     float* __restrict__ ys,          // (B*L,512)
                     int L)
{
    const int blk  = blockIdx.x;      // b*512 + d
    const int b    = blk >> 9;
    const int d    = blk & 511;
    const int lane = threadIdx.x;     // 0..31
    const float A0 = Aneg[d * 64 + lane];
    const float A1 = Aneg[d * 64 + lane + 32];
    float h0 = 0.0f, h1 = 0.0f;

    #pragma unroll 1
    for (int l = 0; l < L; ++l) {
        const size_t row = (size_t)b * L + l;
        const float dt = delta[row * 512 + d];
        const float uu = u[row * 512 + d];
        const float* xr = xdbl + row * 144;
        const float Bv0 = xr[16 + lane], Bv1 = xr[48 + lane];
        const float Cv0 = xr[80 + lane], Cv1 = xr[112 + lane];
        const float du = dt * uu;
        h0 = fmaf(h0, __expf(dt * A0), du * Bv0);
        h1 = fmaf(h1, __expf(dt * A1), du * Bv1);
        float p = fmaf(h0, Cv0, h1 * Cv1);
        #pragma unroll
        for (int off = 16; off > 0; off >>= 1) p += __shfl_xor(p, off, 32);
        if (lane == 0) ys[row * 512 + d] = p;
    }
}

// ys[i] = (ys[i] + u[i]*D[d]) * silu(z[i]),  z = xz cols [512,1024)
__global__ void gate_kernel(float* __restrict__ ys, const float* __restrict__ u,
                            const float* __restrict__ D, const float* __restrict__ xz,
                            int M)
{
    int idx = blockIdx.x * blockDim.x + threadIdx.x;
    if (idx >= M * 512) return;
    int d = idx & 511; int m = idx >> 9;
    float z = xz[(size_t)m * 1024 + 512 + d];
    ys[idx] = (ys[idx] + u[idx] * D[d]) * siluf_(z);
}

// dst(B,C,64,64) += bilinear_upsample(src(B,C,16,16))  (half-pixel centers)
__global__ void upsample_add_kernel(const float* __restrict__ src, float* __restrict__ dst,
                                    int B, int C)
{
    int idx = blockIdx.x * blockDim.x + threadIdx.x;
    int total = B * C * 64 * 64;
    if (idx >= total) return;
    int x = idx & 63; int t = idx >> 6;
    int y = t & 63;   t >>= 6;
    int c = t % C;    int b = t / C;

    float sx = (x + 0.5f) * 0.25f - 0.5f;
    float sy = (y + 0.5f) * 0.25f - 0.5f;
    int x0 = (int)floorf(sx), y0 = (int)floorf(sy);
    float wx = sx - x0, wy = sy - y0;
    int x0c = min(max(x0, 0), 15), x1c = min(max(x0 + 1, 0), 15);
    int y0c = min(max(y0, 0), 15), y1c = min(max(y0 + 1, 0), 15);

    const float* sp = src + (size_t)(b * C + c) * 256;
    float v00 = sp[y0c * 16 + x0c], v01 = sp[y0c * 16 + x1c];
    float v10 = sp[y1c * 16 + x0c], v11 = sp[y1c * 16 + x1c];
    float v0 = v00 + (v01 - v00) * wx;
    float v1 = v10 + (v11 - v10) * wx;
    dst[idx] += v0 + (v1 - v0) * wy;
}

// ===========================================================================
// Host launcher
// ===========================================================================
#define CDIV(a, b) (((a) + (b) - 1) / (b))
#define L1D(kern, total, ...) kern<<<CDIV((total), 256), 256, 0, stream>>>(__VA_ARGS__)

extern "C" void kernel_launch(void* const* d_in, const int* in_sizes, int n_in,
                              void* d_out, int out_size, void* d_ws, size_t ws_size,
                              hipStream_t stream)
{
    (void)in_sizes; (void)n_in; (void)out_size; (void)ws_size;

    const float* x         = (const float*)d_in[0];
    const float* w1        = (const float*)d_in[1];
    const float* b1        = (const float*)d_in[2];
    const float* w2        = (const float*)d_in[3];
    const float* b2        = (const float*)d_in[4];
    const float* w3        = (const float*)d_in[5];
    const float* b3        = (const float*)d_in[6];
    const float* w1s       = (const float*)d_in[7];
    const float* b1s       = (const float*)d_in[8];
    const float* w2s       = (const float*)d_in[9];
    const float* b2s       = (const float*)d_in[10];
    const float* wm        = (const float*)d_in[11];
    const float* bm        = (const float*)d_in[12];
    const float* in_proj_w = (const float*)d_in[13];
    const float* conv1d_w  = (const float*)d_in[14];
    const float* conv1d_b  = (const float*)d_in[15];
    const float* x_proj_w  = (const float*)d_in[16];
    const float* dt_proj_w = (const float*)d_in[17];
    const float* dt_proj_b = (const float*)d_in[18];
    const float* A_log     = (const float*)d_in[19];
    const float* D_ssm     = (const float*)d_in[20];
    const float* out_proj_w= (const float*)d_in[21];
    const float* dw1       = (const float*)d_in[22];
    const float* db1       = (const float*)d_in[23];
    const float* dw2       = (const float*)d_in[24];
    const float* db2       = (const float*)d_in[25];
    const float* dw3       = (const float*)d_in[26];
    const float* db3       = (const float*)d_in[27];
    float* out = (float*)d_out;

    const int B = 4, H = 64, W = 64, L = H * W;      // 4096 tokens per batch
    const int M = B * L;                             // 16384 GEMM rows

    // ---- workspace layout (floats) ----
    float* p    = (float*)d_ws;
    float* x1   = p; p += (size_t)B * 64  * H * W;   // (4,64,64,64)
    float* x2   = p; p += (size_t)B * 128 * H * W;   // (4,128,64,64)
    float* x3   = p; p += (size_t)B * 256 * H * W;   // (4,256,64,64)
    float* x1s  = p; p += (size_t)B * 64  * 32 * 32;
    float* x2s  = p; p += (size_t)B * 128 * 16 * 16;
    float* x2m  = p; p += (size_t)B * 256 * 16 * 16;
    float* x3f  = p; p += (size_t)M * 256;           // (b,l,256)
    float* xz   = p; p += (size_t)M * 1024;          // in_proj out
    float* u    = p; p += (size_t)M * 512;           // conv1d+silu
    float* xdbl = p; p += (size_t)M * 144;           // x_proj out (dt|B|C)
    float* delta= p; p += (size_t)M * 512;
    float* Aneg = p; p += (size_t)512 * 64;
    float* ys   = p; p += (size_t)M * 512;           // scan out, gated in place
    float* y2   = p; p += (size_t)M * 256;           // out_proj
    float* x3s  = p; p += (size_t)B * 256 * H * W;   // (4,256,64,64)
    float* t2   = p; p += (size_t)B * 128 * H * W;
    float* t3   = p; p += (size_t)B * 64  * H * W;

    // ---- encoder convs ----
    L1D(conv2d_kernel, B*64*H*W,  x,   w1,  b1,  x1,  B,   3, 64, 64,  64, 64, 64, 3,3, 1,1, 1, 0);
    L1D(conv2d_kernel, B*128*H*W, x1,  w2,  b2,  x2,  B,  64, 64, 64, 128, 64, 64, 3,3, 1,1, 1, 0);
    L1D(conv2d_kernel, B*256*H*W, x2,  w3,  b3,  x3,  B, 128, 64, 64, 256, 64, 64, 3,3, 1,1, 1, 0);
    L1D(conv2d_kernel, B*64*32*32,  x,   w1s, b1s, x1s, B,   3, 64, 64,  64, 32, 32, 3,3, 2,1, 1, 0);
    L1D(conv2d_kernel, B*128*16*16, x1s, w2s, b2s, x2s, B,  64, 32, 32, 128, 16, 16, 3,3, 2,1, 1, 0);
    L1D(conv2d_kernel, B*256*16*16, x2s, wm,  bm,  x2m, B, 128, 16, 16, 256, 16, 16, 1,1, 1,0, 0, 0);

    // ---- Mamba ----
    L1D(chw_to_lc_kernel, B*256*L, x3, x3f, B, 256, L);

    // xz = x3f @ in_proj_w^T : (16384,256)x(1024,256)^T  [WMMA strip]
    {
        int waves = (M / 16) * (1024 / 64);
        gemm_wmma_f32_strip_kernel<<<CDIV(waves, 4), 128, 0, stream>>>(
            x3f, 256, in_proj_w, nullptr, xz, 1024, M, 1024, 256);
    }

    // u = silu(causal depthwise conv1d(xs))  [TDM weight staging]
    L1D(conv1d_silu_kernel, M*512, xz, conv1d_w, conv1d_b, u, B, L);

    // x_dbl = u @ x_proj_w^T : (16384,512)x(144,512)^T  [WMMA 16x16]
    {
        int waves = (M / 16) * (144 / 16);
        gemm_wmma_f32_kernel<<<CDIV(waves, 4), 128, 0, stream>>>(
            u, 512, x_proj_w, nullptr, xdbl, 144, M, 144, 512);
    }

    // delta = softplus(dt @ dt_proj_w^T + b) : A = x_dbl[:, :16] (lda=144) [WMMA strip]
    {
        int waves = (M / 16) * (512 / 64);
        gemm_wmma_f32_strip_kernel<<<CDIV(waves, 4), 128, 0, stream>>>(
            xdbl, 144, dt_proj_w, nullptr, delta, 512, M, 512, 16);
    }
    L1D(bias_softplus_kernel, M*512, delta, dt_proj_b, M, 512);

    L1D(neg_exp_kernel, 512*64, A_log, Aneg, 512*64);

    // selective scan: 2048 waves, one per (b,d)
    ssm_scan_kernel<<<B * 512, 32, 0, stream>>>(u, delta, xdbl, Aneg, ys, L);

    // y = (ys + u*D) * silu(z)   (in place on ys)
    L1D(gate_kernel, M*512, ys, u, D_ssm, xz, M);

    // y2 = y @ out_proj_w^T : (16384,512)x(256,512)^T  [WMMA strip]
    {
        int waves = (M / 16) * (256 / 64);
        gemm_wmma_f32_strip_kernel<<<CDIV(waves, 4), 128, 0, stream>>>(
            ys, 512, out_proj_w, nullptr, y2, 256, M, 256, 512);
    }

    L1D(lc_to_chw_kernel, B*256*L, y2, x3s, B, 256, L);

    // ---- decoder ----
    L1D(upsample_add_kernel, B*256*H*W, x2m, x3s, B, 256);                    // x3s += up(x2m)
    L1D(conv2d_kernel, B*128*H*W, x3s, dw1, db1, t2, B, 256, 64, 64, 128, 64, 64, 3,3, 1,1, 1, 1);
    L1D(add_inplace_kernel, B*128*H*W, t2, x2, B*128*H*W);
    L1D(conv2d_kernel, B*64*H*W,  t2,  dw2, db2, t3, B, 128, 64, 64,  64, 64, 64, 3,3, 1,1, 1, 1);
    L1D(add_inplace_kernel, B*64*H*W, t3, x1, B*64*H*W);
    L1D(conv2d_kernel, B*3*H*W,   t3,  dw3, db3, out, B, 64, 64, 64,   3, 64, 64, 3,3, 1,1, 2, 1);
}